// SelectiveSSM_1855425872531
// MI455X (gfx1250) — compile-verified
//
#include <hip/hip_runtime.h>
#include <hip/hip_bf16.h>
#include <math.h>

// ---------------------------------------------------------------------------
// Mamba selective-SSM block for gfx1250 (MI455X), fp32 via V_WMMA_F32_16X16X4_F32
// LDS-tiled GEMMs (double-buffered), gfx1250 async global->LDS copies.
// ---------------------------------------------------------------------------

#define D_MODEL 1024
#define D_STATE 16
#define D_CONV  4
#define DT_RANK 64
#define D_INNER 2048
#define BATCH   2
#define SEQ_LEN 2048
#define ROWS    (BATCH * SEQ_LEN)      // 4096 token rows

typedef float v2f __attribute__((ext_vector_type(2)));
typedef float v8f __attribute__((ext_vector_type(8)));
typedef int   v4i __attribute__((vector_size(16)));   // b128 payload type

// gfx1250 async global->LDS copy path (guarded; falls back to ld/ds_store)
#if defined(__gfx1250__) && __has_builtin(__builtin_amdgcn_global_load_async_to_lds_b128) && \
    __has_builtin(__builtin_amdgcn_s_wait_asynccnt)
#define USE_ASYNC_LDS 1
#define GLB_V4I(p) ((__attribute__((address_space(1))) v4i*)(p))
#define LDS_V4I(p) ((__attribute__((address_space(3))) v4i*)(p))
#else
#define USE_ASYNC_LDS 0
#endif

// Tile geometry: block = 256 threads (8 waves), output 64(M) x 128(N),
// wave grid 2x4, each wave 32x32 = 2x2 WMMA tiles. K staged in chunks of 16.
#define BM 64
#define BN 128
#define KT 16
#define APAD 20   // padded LDS row stride (floats): conflict-free & 16B aligned

template <int MODE>
__global__ __launch_bounds__(256) void wmma_gemm_nt(
    const float* __restrict__ A, int lda,
    const float* __restrict__ B, int ldb,
    float* __restrict__ C, int ldc,
    int M, int N, int K,
    const float* __restrict__ bias)
{
    __shared__ float Asm[2][BM * APAD];   // 2 x 5120 B
    __shared__ float Bsm[2][BN * APAD];   // 2 x 10240 B

    const int tid  = threadIdx.x;
    const int lane = tid & 31;
    const int wave = tid >> 5;            // 0..7
    const int wm   = wave >> 2;           // 0..1  (M direction)
    const int wn   = wave & 3;            // 0..3  (N direction)
    const int m0   = blockIdx.x * BM;
    const int n0   = blockIdx.y * BN;

    const int r    = lane & 15;
    const int half = lane >> 4;           // 0: K={0,1}, 1: K={2,3}

    // wave-uniform compute guard (keeps EXEC all-1s for WMMA; never skips barriers)
    const bool wactive = (n0 + wn * 32) < N;

    // staging indices: A = 64x16 (256 float4), B = 128x16 (512 float4, 2/thread)
    const int arow = tid >> 2;            // 0..63
    const int ac4  = (tid & 3) * 4;
    const int brow0 = tid >> 2;           // rows 0..63   (f = tid)
    const int bc40  = (tid & 3) * 4;
    const int brow1 = (tid + 256) >> 2;   // rows 64..127 (f = tid + 256)
    const int bc41  = (tid & 3) * 4;

    const int nk = K / KT;

#if USE_ASYNC_LDS
    auto stage_issue = [&](int buf, int k) {
        __builtin_amdgcn_global_load_async_to_lds_b128(
            GLB_V4I(A + (size_t)(m0 + arow) * lda + k + ac4),
            LDS_V4I(&Asm[buf][arow * APAD + ac4]), 0, 0);
        if (n0 + brow0 < N)
            __builtin_amdgcn_global_load_async_to_lds_b128(
                GLB_V4I(B + (size_t)(n0 + brow0) * ldb + k + bc40),
                LDS_V4I(&Bsm[buf][brow0 * APAD + bc40]), 0, 0);
        if (n0 + brow1 < N)
            __builtin_amdgcn_global_load_async_to_lds_b128(
                GLB_V4I(B + (size_t)(n0 + brow1) * ldb + k + bc41),
                LDS_V4I(&Bsm[buf][brow1 * APAD + bc41]), 0, 0);
    };
    auto stage_commit = [&](int) {};
    auto stage_wait   = [&]() { __builtin_amdgcn_s_wait_asynccnt(0); };
#else
    float4 ra, rb0, rb1;
    auto stage_issue = [&](int, int k) {
        ra = *(const float4*)(A + (size_t)(m0 + arow) * lda + k + ac4);
        rb0 = (n0 + brow0 < N) ? *(const float4*)(B + (size_t)(n0 + brow0) * ldb + k + bc40)
                               : float4{0.f, 0.f, 0.f, 0.f};
        rb1 = (n0 + brow1 < N) ? *(const float4*)(B + (size_t)(n0 + brow1) * ldb + k + bc41)
                               : float4{0.f, 0.f, 0.f, 0.f};
    };
    auto stage_commit = [&](int buf) {
        *(float4*)&Asm[buf][arow * APAD + ac4]   = ra;
        *(float4*)&Bsm[buf][brow0 * APAD + bc40] = rb0;
        *(float4*)&Bsm[buf][brow1 * APAD + bc41] = rb1;
    };
    auto stage_wait = [&]() {};
#endif

    v8f acc[2][2] = {};

    // prologue: stage k-chunk 0 into buffer 0
    stage_issue(0, 0);
    stage_commit(0);
    stage_wait();
    __syncthreads();

    for (int kc = 0; kc < nk; ++kc) {
        const int cur = kc & 1;
        if (kc + 1 < nk) stage_issue(cur ^ 1, (kc + 1) * KT);

        if (wactive) {
            const float* As = &Asm[cur][(wm * 32) * APAD];
            const float* Bs = &Bsm[cur][(wn * 32) * APAD];
#pragma unroll
            for (int kk = 0; kk < KT; kk += 4) {
                v2f a0 = *(const v2f*)&As[r * APAD + kk + 2 * half];
                v2f a1 = *(const v2f*)&As[(16 + r) * APAD + kk + 2 * half];
                v2f b0 = *(const v2f*)&Bs[r * APAD + kk + 2 * half];
                v2f b1 = *(const v2f*)&Bs[(16 + r) * APAD + kk + 2 * half];
                acc[0][0] = __builtin_amdgcn_wmma_f32_16x16x4_f32(false, a0, false, b0, (short)0, acc[0][0], false, false);
                acc[0][1] = __builtin_amdgcn_wmma_f32_16x16x4_f32(false, a0, false, b1, (short)0, acc[0][1], false, false);
                acc[1][0] = __builtin_amdgcn_wmma_f32_16x16x4_f32(false, a1, false, b0, (short)0, acc[1][0], false, false);
                acc[1][1] = __builtin_amdgcn_wmma_f32_16x16x4_f32(false, a1, false, b1, (short)0, acc[1][1], false, false);
            }
        }

        if (kc + 1 < nk) stage_commit(cur ^ 1);
        stage_wait();
        __syncthreads();
    }

    if (wactive) {
#pragma unroll
        for (int i2 = 0; i2 < 2; ++i2) {
#pragma unroll
            for (int j2 = 0; j2 < 2; ++j2) {
                const int col = n0 + wn * 32 + j2 * 16 + r;
                if (col < N) {
#pragma unroll
                    for (int i = 0; i < 8; ++i) {
                        const int row = m0 + wm * 32 + i2 * 16 + i + 8 * half;
                        float v = acc[i2][j2][i];
                        if (MODE == 1) {                 // softplus(v + bias)
                            v += bias[col];
                            v = (v > 20.0f) ? v : log1pf(__expf(v));
                        }
                        C[(size_t)row * ldc + col] = v;
                    }
                }
            }
        }
    }
}

// ---------------------------------------------------------------------------
// Depthwise causal conv1d (K=4, left pad 3) over the xc half of xz.
// ---------------------------------------------------------------------------
__global__ __launch_bounds__(256) void conv1d_kernel(
    const float* __restrict__ xz,
    const float* __restrict__ conv_w,   // [D_INNER, 1, 4]
    const float* __restrict__ conv_b,   // [D_INNER]
    float* __restrict__ xs)
{
    const size_t idx = (size_t)blockIdx.x * blockDim.x + threadIdx.x;
    if (idx >= (size_t)ROWS * D_INNER) return;
    const int d  = (int)(idx & (D_INNER - 1));
    const int ml = (int)(idx >> 11);            // b*SEQ_LEN + l
    const int l  = ml & (SEQ_LEN - 1);

    const float w0 = conv_w[d * 4 + 0];
    const float w1 = conv_w[d * 4 + 1];
    const float w2 = conv_w[d * 4 + 2];
    const float w3 = conv_w[d * 4 + 3];

    float acc = conv_b[d];
    const size_t base = (size_t)ml * (2 * D_INNER) + d;
    const size_t rstr = 2 * D_INNER;
    if (l >= 3) acc += w0 * xz[base - 3 * rstr];
    if (l >= 2) acc += w1 * xz[base - 2 * rstr];
    if (l >= 1) acc += w2 * xz[base - 1 * rstr];
    acc += w3 * xz[base];
    xs[idx] = acc;
}

// ---------------------------------------------------------------------------
// Selective scan, lane-per-state: 16 lanes = 16 states of one (b,d) channel.
// Fuses skip term (D*x) and SiLU gate; gated y overwrites the dt buffer.
// ---------------------------------------------------------------------------
__global__ __launch_bounds__(256) void scan_kernel(
    const float* __restrict__ xz,       // z at cols [D_INNER, 2*D_INNER)
    const float* __restrict__ xs,       // [ROWS, D_INNER]
    const float* __restrict__ x_dbl,    // [ROWS, 96]; B at [64,80), C at [80,96)
    float* __restrict__ dt_y,           // in: dt [ROWS, D_INNER]; out: gated y
    const float* __restrict__ A_log,    // [D_INNER, D_STATE]
    const float* __restrict__ Dp)       // [D_INNER]
{
    const int tid = blockIdx.x * blockDim.x + threadIdx.x;   // 0 .. 65535
    const int n   = tid & 15;                                // state index
    const int ch  = tid >> 4;                                // 0 .. 4095
    const int b   = ch >> 11;
    const int d   = ch & (D_INNER - 1);

    const float Aval = -__expf(A_log[d * D_STATE + n]);
    const float Dd   = Dp[d];

    float h = 0.0f;
    for (int l = 0; l < SEQ_LEN; ++l) {
        const int    row = b * SEQ_LEN + l;
        const size_t rb  = (size_t)row * D_INNER + d;
        const float dt_v = dt_y[rb];                         // broadcast in group
        const float x_v  = xs[rb];
        const float Bn   = x_dbl[(size_t)row * 96 + DT_RANK + n];
        const float Cn   = x_dbl[(size_t)row * 96 + DT_RANK + D_STATE + n];

        if (l + 8 < SEQ_LEN) {                               // global_prefetch_b8
            __builtin_prefetch(&dt_y[rb + 8 * D_INNER], 0, 3);
            __builtin_prefetch(&xs[rb + 8 * D_INNER], 0, 3);
            __builtin_prefetch(&x_dbl[(size_t)(row + 8) * 96 + DT_RANK + n], 0, 3);
        }

        h = __expf(dt_v * Aval) * h + dt_v * x_v * Bn;
        float part = h * Cn;
        // 16-lane tree reduction (masks stay inside each 16-lane group)
        part += __shfl_xor(part, 8, 32);
        part += __shfl_xor(part, 4, 32);
        part += __shfl_xor(part, 2, 32);
        part += __shfl_xor(part, 1, 32);

        if (n == 0) {
            const float zv = xz[(size_t)row * (2 * D_INNER) + D_INNER + d];
            const float y  = part + Dd * x_v;
            dt_y[rb] = y * (zv / (1.0f + __expf(-zv)));      // y * SiLU(z)
        }
    }
}

// ---------------------------------------------------------------------------
// Launch
// ---------------------------------------------------------------------------
extern "C" void kernel_launch(void* const* d_in, const int* in_sizes, int n_in,
                              void* d_out, int out_size, void* d_ws, size_t ws_size,
                              hipStream_t stream)
{
    const float* x         = (const float*)d_in[0];   // [2,2048,1024]
    const float* in_proj_w = (const float*)d_in[1];   // [4096,1024]
    const float* conv_w    = (const float*)d_in[2];   // [2048,1,4]
    const float* conv_b    = (const float*)d_in[3];   // [2048]
    const float* A_log     = (const float*)d_in[4];   // [2048,16]
    const float* x_proj_w  = (const float*)d_in[5];   // [96,2048]
    const float* dt_proj_w = (const float*)d_in[6];   // [2048,64]
    const float* dt_proj_b = (const float*)d_in[7];   // [2048]
    const float* D_param   = (const float*)d_in[8];   // [2048]
    const float* out_proj_w= (const float*)d_in[9];   // [1024,2048]
    float* out = (float*)d_out;                       // [2,2048,1024]

    float* ws    = (float*)d_ws;
    float* xz    = ws;                                   // 4096 x 4096
    float* xs    = xz  + (size_t)ROWS * 2 * D_INNER;     // 4096 x 2048
    float* x_dbl = xs  + (size_t)ROWS * D_INNER;         // 4096 x 96
    float* dt_y  = x_dbl + (size_t)ROWS * 96;            // 4096 x 2048 (dt, then y)

    // 1) xz = x @ in_proj_w^T              [4096 x 4096, K=1024]
    wmma_gemm_nt<0><<<dim3(ROWS / BM, (2 * D_INNER) / BN), 256, 0, stream>>>(
        x, D_MODEL, in_proj_w, D_MODEL, xz, 2 * D_INNER,
        ROWS, 2 * D_INNER, D_MODEL, nullptr);

    // 2) depthwise causal conv -> xs
    conv1d_kernel<<<(ROWS * D_INNER) / 256, 256, 0, stream>>>(xz, conv_w, conv_b, xs);

    // 3) x_dbl = xs @ x_proj_w^T           [4096 x 96, K=2048]
    wmma_gemm_nt<0><<<dim3(ROWS / BM, 1), 256, 0, stream>>>(
        xs, D_INNER, x_proj_w, D_INNER, x_dbl, 96,
        ROWS, 96, D_INNER, nullptr);

    // 4) dt = softplus(x_dbl[:, :64] @ dt_proj_w^T + b)   [4096 x 2048, K=64]
    wmma_gemm_nt<1><<<dim3(ROWS / BM, D_INNER / BN), 256, 0, stream>>>(
        x_dbl, 96, dt_proj_w, DT_RANK, dt_y, D_INNER,
        ROWS, D_INNER, DT_RANK, dt_proj_b);

    // 5) selective scan + skip + SiLU gate (writes gated y into dt_y)
    scan_kernel<<<(BATCH * D_INNER * D_STATE) / 256, 256, 0, stream>>>(
        xz, xs, x_dbl, dt_y, A_log, D_param);

    // 6) out = y @ out_proj_w^T            [4096 x 1024, K=2048]
    wmma_gemm_nt<0><<<dim3(ROWS / BM, D_MODEL / BN), 256, 0, stream>>>(
        dt_y, D_INNER, out_proj_w, D_INNER, out, D_MODEL,
        ROWS, D_MODEL, D_INNER, nullptr);
}